// LuongAttention_85753317032704
// MI455X (gfx1250) — compile-verified
//
#include <hip/hip_runtime.h>
#include <stdint.h>

#define B_ 64
#define S_ 4096
#define H_ 1024
#define CHUNKS 16
#define ROWS_PER_CHUNK (S_ / CHUNKS)      // 256
#define WAVES 8
#define BLOCK 256
#define TILE_ROWS 8                        // one row per wave per tile
#define TILES (ROWS_PER_CHUNK / TILE_ROWS) // 32
#define NEG_INF_F (-1.0e9f)

typedef float v4f __attribute__((ext_vector_type(4)));
typedef float v2f __attribute__((ext_vector_type(2)));
typedef float v8f __attribute__((ext_vector_type(8)));

// CDNA5 async global->LDS copy (ASYNCcnt-tracked). VDST = LDS byte address,
// VADDR = 32-bit byte offset, SADDR = 64-bit base (GVS mode).
__device__ __forceinline__ void async_ld_b128(uint32_t lds_byte, uint32_t goff_byte,
                                              uint64_t gbase) {
  asm volatile("global_load_async_to_lds_b128 %0, %1, %2 offset:0"
               :: "v"(lds_byte), "v"(goff_byte), "s"(gbase)
               : "memory");
}
__device__ __forceinline__ void wait_async_le8() {
  asm volatile("s_wait_asynccnt 0x8" ::: "memory");
}
__device__ __forceinline__ void wait_async_0() {
  asm volatile("s_wait_asynccnt 0x0" ::: "memory");
}

// ---------------------------------------------------------------------------
// Pass 1: streaming flash pass. One workgroup = (batch b, chunk g of 256 rows).
// Double-buffered async tiles (8 rows x 1024 f32 = 32 KB) in LDS; each wave
// owns one row per tile: dot with hidden (kept in regs), online softmax,
// context accumulation from the same registers (E read from HBM exactly once).
// ---------------------------------------------------------------------------
__global__ __launch_bounds__(BLOCK) void attn_partial(
    const float* __restrict__ hidden, const float* __restrict__ enc,
    const int* __restrict__ mask, float* __restrict__ scores_out,
    float* __restrict__ mpart, float* __restrict__ lpart,
    float* __restrict__ ctxpart) {
  __shared__ __align__(16) float ldsE[2][TILE_ROWS * H_];   // 64 KB
  __shared__ float s_m[WAVES], s_l[WAVES];

  const int g = blockIdx.x;
  const int b = blockIdx.y;
  const int tid = threadIdx.x;
  const int lane = tid & 31;
  const int wv = tid >> 5;

  const float* gsrc = enc + ((size_t)b * S_ + (size_t)g * ROWS_PER_CHUNK) * H_;
  const uint64_t gbase = (uint64_t)(uintptr_t)gsrc;
  const uint32_t lds0 = (uint32_t)(uintptr_t)(&ldsE[0][0]);  // low 32 bits = LDS offset

  // Per-lane slice of decoder_hidden: h = lane*4 + c + j*128 (matches E tile layout)
  v4f hreg[8];
#pragma unroll
  for (int j = 0; j < 8; ++j)
    hreg[j] = *(const v4f*)(hidden + (size_t)b * H_ + lane * 4 + j * 128);

  float ctx[8][4];
#pragma unroll
  for (int j = 0; j < 8; ++j)
#pragma unroll
    for (int c = 0; c < 4; ++c) ctx[j][c] = 0.0f;
  float m = -3.0e38f, l = 0.0f;

  // Prefetch tile 0 into buffer 0 (8 b128 async ops per thread per tile)
#pragma unroll
  for (int j = 0; j < 8; ++j) {
    const uint32_t f = (uint32_t)(j * 4096 + tid * 16);
    async_ld_b128(lds0 + f, f, gbase);
  }

  for (int t = 0; t < TILES; ++t) {
    const int buf = t & 1;
    if (t + 1 < TILES) {
      const uint32_t go = (uint32_t)((t + 1) * TILE_ROWS * H_ * 4);
      const uint32_t lo = lds0 + (uint32_t)((buf ^ 1) * TILE_ROWS * H_ * 4);
#pragma unroll
      for (int j = 0; j < 8; ++j) {
        const uint32_t f = (uint32_t)(j * 4096 + tid * 16);
        async_ld_b128(lo + f, go + f, gbase);
      }
      wait_async_le8();   // >8 outstanding drained => tile t complete (in-order)
    } else {
      wait_async_0();
    }
    __syncthreads();       // tile t visible to all waves

    // ---- wave wv processes row wv of tile t ----
    const float* row = &ldsE[buf][wv * H_];
    v4f e[8];
#pragma unroll
    for (int j = 0; j < 8; ++j)
      e[j] = *(const v4f*)(row + lane * 4 + j * 128);

    float acc = 0.0f;
#pragma unroll
    for (int j = 0; j < 8; ++j)
#pragma unroll
      for (int c = 0; c < 4; ++c) acc = fmaf(e[j][c], hreg[j][c], acc);

#pragma unroll
    for (int off = 16; off > 0; off >>= 1) acc += __shfl_xor(acc, off);

    const int s = g * ROWS_PER_CHUNK + t * TILE_ROWS + wv;
    const float sc = mask[(size_t)b * S_ + s] ? acc : NEG_INF_F;
    if (lane == 0) scores_out[(size_t)b * S_ + s] = sc;  // raw masked score

    // online softmax update + context accumulation from the held registers
    const float mn = fmaxf(m, sc);
    const float corr = __expf(m - mn);
    const float p = __expf(sc - mn);
    l = l * corr + p;
#pragma unroll
    for (int j = 0; j < 8; ++j)
#pragma unroll
      for (int c = 0; c < 4; ++c)
        ctx[j][c] = fmaf(ctx[j][c], corr, p * e[j][c]);
    m = mn;

    __syncthreads();       // reads done before this buffer is overwritten
  }

  // ---- intra-workgroup combine: 8 wave partials -> 1 chunk partial ----
  if (lane == 0) { s_m[wv] = m; s_l[wv] = l; }
  float* cacc = &ldsE[0][0];  // reuse tile buffer (all reads complete)
  for (int q = tid; q < H_; q += BLOCK) cacc[q] = 0.0f;
  __syncthreads();

  float Mc = s_m[0];
#pragma unroll
  for (int i = 1; i < WAVES; ++i) Mc = fmaxf(Mc, s_m[i]);
  float Lc = 0.0f;
#pragma unroll
  for (int i = 0; i < WAVES; ++i) Lc += __expf(s_m[i] - Mc) * s_l[i];
  const float coef = __expf(m - Mc);

#pragma unroll
  for (int j = 0; j < 8; ++j)
#pragma unroll
    for (int c = 0; c < 4; ++c)
      atomicAdd(&cacc[lane * 4 + c + j * 128], coef * ctx[j][c]);  // ds_add_f32
  __syncthreads();

  float* cp = ctxpart + ((size_t)b * CHUNKS + g) * H_;
  for (int q = tid; q < H_; q += BLOCK) cp[q] = cacc[q];
  if (tid == 0) {
    mpart[b * CHUNKS + g] = Mc;
    lpart[b * CHUNKS + g] = Lc;
  }
}

// ---------------------------------------------------------------------------
// Pass 2: per batch, reduce the 16 chunk partials and normalize.
// Context combine uses V_WMMA_F32_16X16X4_F32: A = coef broadcast (16x4),
// B[k,n] = ctxpart[group k][h0+n], chained over 4 WMMAs (K=16 groups).
// D rows are identical; row 0 (VGPR0, lanes 0-15) gives the combined context.
// ---------------------------------------------------------------------------
__global__ __launch_bounds__(BLOCK) void attn_combine(
    const float* __restrict__ mpart, const float* __restrict__ lpart,
    const float* __restrict__ ctxpart, float* __restrict__ ctx_out,
    float* __restrict__ w_out) {
  __shared__ float s_coef[CHUNKS];
  __shared__ float s_M, s_invL;

  const int b = blockIdx.x;
  const int tid = threadIdx.x;
  const int lane = tid & 31;
  const int wv = tid >> 5;

  if (wv == 0) {
    float mg = (lane < CHUNKS) ? mpart[b * CHUNKS + lane] : -3.0e38f;
    float M = mg;
#pragma unroll
    for (int off = 16; off > 0; off >>= 1) M = fmaxf(M, __shfl_xor(M, off));
    float lg = (lane < CHUNKS) ? lpart[b * CHUNKS + lane] : 0.0f;
    float cg = (lane < CHUNKS) ? __expf(mg - M) : 0.0f;
    float Ls = cg * lg;
#pragma unroll
    for (int off = 16; off > 0; off >>= 1) Ls += __shfl_xor(Ls, off);
    if (lane < CHUNKS) s_coef[lane] = cg;
    if (lane == 0) { s_M = M; s_invL = 1.0f / Ls; }
  }
  __syncthreads();

  const int lane16 = lane & 15;
  const int hi2 = (lane >= 16) ? 2 : 0;   // upper half-wave holds K=2,3 slices
  const float invL = s_invL;

  // 64 h-tiles of 16, 8 per wave. EXEC all-ones throughout (selects, no branches).
  for (int i = 0; i < 8; ++i) {
    const int h0 = (wv * 8 + i) * 16;
    v8f c = {0.f, 0.f, 0.f, 0.f, 0.f, 0.f, 0.f, 0.f};
#pragma unroll
    for (int kc = 0; kc < 4; ++kc) {
      v2f a, bb;
      a.x = s_coef[kc * 4 + hi2];          // A[m,K=hi2]   (all m rows equal)
      a.y = s_coef[kc * 4 + hi2 + 1];      // A[m,K=hi2+1]
      const float* bp =
          ctxpart + ((size_t)b * CHUNKS + kc * 4 + hi2) * H_ + h0 + lane16;
      bb.x = bp[0];                        // B[K=hi2,   n=lane16]
      bb.y = bp[H_];                       // B[K=hi2+1, n=lane16]
      c = __builtin_amdgcn_wmma_f32_16x16x4_f32(false, a, false, bb, (short)0, c,
                                                false, false);
    }
    if (lane < 16)
      ctx_out[(size_t)b * H_ + h0 + lane16] = c[0] * invL;  // D row M=0
  }

  // scores -> normalized attention weights (in place in output region)
  const float M = s_M;
  for (int idx = tid; idx < S_; idx += BLOCK) {
    const float sc = w_out[(size_t)b * S_ + idx];
    w_out[(size_t)b * S_ + idx] = __expf(sc - M) * invL;
  }
}

// ---------------------------------------------------------------------------
extern "C" void kernel_launch(void* const* d_in, const int* in_sizes, int n_in,
                              void* d_out, int out_size, void* d_ws, size_t ws_size,
                              hipStream_t stream) {
  (void)in_sizes; (void)n_in; (void)out_size; (void)ws_size;
  const float* hidden = (const float*)d_in[0];          // [B,H] f32
  const float* enc = (const float*)d_in[1];             // [B,S,H] f32
  const int* mask = (const int*)d_in[2];                // [B,S] bool->int

  float* ctx_out = (float*)d_out;                       // [B,H]
  float* w_out = ctx_out + (size_t)B_ * H_;             // [B,S] (scores then weights)

  float* mpart = (float*)d_ws;                          // [B,16]
  float* lpart = mpart + B_ * CHUNKS;                   // [B,16]
  float* ctxpart = lpart + B_ * CHUNKS;                 // [B,16,H]  ~4 MB

  attn_partial<<<dim3(CHUNKS, B_), BLOCK, 0, stream>>>(hidden, enc, mask, w_out,
                                                       mpart, lpart, ctxpart);
  attn_combine<<<B_, BLOCK, 0, stream>>>(mpart, lpart, ctxpart, ctx_out, w_out);
}